// cyclicLoss_6725918786237
// MI455X (gfx1250) — compile-verified
//
#include <hip/hip_runtime.h>
#include <hip/hip_bf16.h>
#include <math.h>

#define LAMDA 0.1f

typedef __attribute__((ext_vector_type(2))) float v2f;
typedef __attribute__((ext_vector_type(4))) float v4f;
typedef __attribute__((ext_vector_type(8))) float v8f;

// Wave32 full reduction using the matrix unit as a row-sum engine.
// A (16x4 f32): lane L<16 -> row M=L holds {K0=acc, K1=0}; lane L>=16 -> row M=L-16 holds {K2=acc, K3=0}.
// B = all ones  =>  D[m][n] = acc[m] + acc[m+16] for every column n.
// Each lane sums its 8 D VGPRs (rows 0..7 or 8..15), then one xor-16 shuffle gives the wave total.
__device__ __forceinline__ float wave_reduce_f32(float v) {
  v2f a; a[0] = v;    a[1] = 0.0f;
  v2f b; b[0] = 1.0f; b[1] = 1.0f;
  v8f c = {};
  v8f d = __builtin_amdgcn_wmma_f32_16x16x4_f32(false, a, false, b, (short)0, c, false, false);
  float s = ((d[0] + d[1]) + (d[2] + d[3])) + ((d[4] + d[5]) + (d[6] + d[7]));
  s += __shfl_xor(s, 16, 32);
  return s;
}

// Block-level reduction of two accumulators (8 waves, 256 threads). Result valid on thread 0.
__device__ __forceinline__ void block_reduce2(float a0, float a1, float* r0, float* r1) {
  __shared__ float s0[8];
  __shared__ float s1[8];
  const int lane = threadIdx.x & 31;
  const int wid  = threadIdx.x >> 5;
  float w0 = wave_reduce_f32(a0);
  float w1 = wave_reduce_f32(a1);
  if (lane == 0) { s0[wid] = w0; s1[wid] = w1; }
  __syncthreads();
  if (threadIdx.x == 0) {
    float t0 = 0.0f, t1 = 0.0f;
    const int nw = blockDim.x >> 5;
    for (int i = 0; i < nw; ++i) { t0 += s0[i]; t1 += s1[i]; }
    *r0 = t0; *r1 = t1;
  }
}

// Stage 1: streaming partial sums. acc0 = sum (in[i]-in[i+T])^2 for i < (N-1)*T,
//          acc1 = sum (in[i]-tg[i])^2. Covers the first n4*4 elements with b128 loads.
__global__ void k_partials(const float* __restrict__ in, const float* __restrict__ tg,
                           const int* __restrict__ pT, const int* __restrict__ pN,
                           int n4, float* __restrict__ ws0, float* __restrict__ ws1) {
  const int T = *pT;
  const int N = *pN;
  const int limit = (N - 1) * T;
  float acc0 = 0.0f, acc1 = 0.0f;
  const int gid    = blockIdx.x * blockDim.x + threadIdx.x;
  const int stride = gridDim.x * blockDim.x;

  if ((T & 3) == 0) {
    const v4f* __restrict__ in4 = (const v4f*)in;
    const v4f* __restrict__ tg4 = (const v4f*)tg;
    const int T4 = T >> 2;
    const int limit4 = limit >> 2;
    for (int i = gid; i < n4; i += stride) {
      v4f a = in4[i];                                    // RT: +T reuse stays in 192MB L2
      v4f t = __builtin_nontemporal_load(&tg4[i]);       // NT: target is read exactly once
      float ex = a.x - t.x, ey = a.y - t.y, ez = a.z - t.z, ew = a.w - t.w;
      acc1 += ex * ex + ey * ey + ez * ez + ew * ew;
      if (i < limit4) {
        v4f b = in4[i + T4];
        float dx = a.x - b.x, dy = a.y - b.y, dz = a.z - b.z, dw = a.w - b.w;
        acc0 += dx * dx + dy * dy + dz * dz + dw * dw;
      }
    }
  } else {  // general fallback if T is not a multiple of 4
    for (int i = gid; i < n4; i += stride) {
#pragma unroll
      for (int j = 0; j < 4; ++j) {
        int idx = i * 4 + j;
        float a = in[idx], t = tg[idx];
        float e = a - t; acc1 += e * e;
        if (idx < limit) { float d = a - in[idx + T]; acc0 += d * d; }
      }
    }
  }

  float b0, b1;
  block_reduce2(acc0, acc1, &b0, &b1);
  if (threadIdx.x == 0) { ws0[blockIdx.x] = b0; ws1[blockIdx.x] = b1; }
}

// Stage 2: single block. Reduce the per-block partials, add the scalar tail
// (elements n4*4 .. n-1), and emit mse + lamda*sqrt(sum0).
__global__ void k_final(const float* __restrict__ ws0, const float* __restrict__ ws1, int nParts,
                        const float* __restrict__ in, const float* __restrict__ tg,
                        const int* __restrict__ pT, const int* __restrict__ pN,
                        int n, int n4, float* __restrict__ out) {
  float a0 = 0.0f, a1 = 0.0f;
  for (int i = threadIdx.x; i < nParts; i += blockDim.x) { a0 += ws0[i]; a1 += ws1[i]; }
  float b0, b1;
  block_reduce2(a0, a1, &b0, &b1);
  if (threadIdx.x == 0) {
    const int T = *pT;
    const int N = *pN;
    const int limit = (N - 1) * T;
    for (int idx = n4 * 4; idx < n; ++idx) {
      float a = in[idx], t = tg[idx];
      float e = a - t; b1 += e * e;
      if (idx < limit) { float d = a - in[idx + T]; b0 += d * d; }
    }
    float mse = b1 / (float)n;
    float nom = (N != 1) ? (LAMDA * sqrtf(b0)) : 0.0f;
    out[0] = mse + nom;
  }
}

extern "C" void kernel_launch(void* const* d_in, const int* in_sizes, int n_in,
                              void* d_out, int out_size, void* d_ws, size_t ws_size,
                              hipStream_t stream) {
  const float* in = (const float*)d_in[0];
  const float* tg = (const float*)d_in[1];
  const int*   pT = (const int*)d_in[2];
  const int*   pN = (const int*)d_in[3];
  float* out = (float*)d_out;

  const int n  = in_sizes[0];
  const int n4 = n >> 2;

  const int threads = 256;                         // 8 wave32 per block
  int blocks = 2048;                               // ~512K threads: enough MLP for 23.3 TB/s
  int needed = (n4 + threads - 1) / threads;
  if (needed < 1) needed = 1;
  if (blocks > needed) blocks = needed;
  size_t wsCap = ws_size / (2 * sizeof(float));
  if (wsCap >= 1 && (size_t)blocks > wsCap) blocks = (int)wsCap;
  if (blocks < 1) blocks = 1;

  float* ws0 = (float*)d_ws;
  float* ws1 = ws0 + blocks;

  k_partials<<<blocks, threads, 0, stream>>>(in, tg, pT, pN, n4, ws0, ws1);
  k_final<<<1, threads, 0, stream>>>(ws0, ws1, blocks, in, tg, pT, pN, n, n4, out);
}